// GCNNet_21912923144343
// MI455X (gfx1250) — compile-verified
//
#include <hip/hip_runtime.h>

#define DIM 128
#define EPS 1e-5f

typedef __attribute__((ext_vector_type(2))) float v2f;
typedef __attribute__((ext_vector_type(8))) float v8f;

// ---------------------------------------------------------------------------
// Degree accumulation: one thread per edge, +1 to out-degree[src], in-degree[dst]
// ---------------------------------------------------------------------------
__global__ void gcn_degree_kernel(const int* __restrict__ src,
                                  const int* __restrict__ dst,
                                  float* __restrict__ deg_out,
                                  float* __restrict__ deg_in, int nedges) {
  int e = blockIdx.x * blockDim.x + threadIdx.x;
  if (e >= nedges) return;
  atomicAdd(&deg_out[src[e]], 1.0f);
  atomicAdd(&deg_in[dst[e]], 1.0f);
}

// in place: d[i] = rsqrt(max(d[i], 1))
__global__ void gcn_norm_kernel(float* __restrict__ d, int n) {
  int i = blockIdx.x * blockDim.x + threadIdx.x;
  if (i >= n) return;
  d[i] = rsqrtf(fmaxf(d[i], 1.0f));
}

// ---------------------------------------------------------------------------
// Tiled GEMM using V_WMMA_F32_16X16X4_F32.
//   out[r, :] = (rowscale ? rowscale[r] : 1) * A[r, :] @ W + bias
// Block = (32, 8): 8 waves. Each block covers a 32-row stripe; wave w owns
// columns [16w, 16w+16) and computes TWO stacked 16x16 tiles (rows 0-15 and
// 16-31 of the stripe) so each B float2 load feeds two WMMAs.
// A stripe (32 x 128) staged in LDS with stride 132: A-operand read address is
// m*132 + k -> bank 4m (and (m+16)*132 adds 2112 = 0 mod 64), conflict-free.
// Safe with out == A (each block fully stages its rows before any store).
// ---------------------------------------------------------------------------
__global__ __launch_bounds__(256) void gcn_gemm_wmma(
    const float* __restrict__ A, const float* __restrict__ W,
    const float* __restrict__ bias, const float* __restrict__ rowscale,
    float* __restrict__ out, int nrows) {
  __shared__ float lA[32 * 132];

  const int tx = threadIdx.x;               // 0..31 lane
  const int wv = threadIdx.y;               // 0..7 wave -> column tile
  const int tid = wv * 32 + tx;             // 0..255
  const int row0 = blockIdx.x * 32;
  const bool full = (row0 + 32) <= nrows;

  // Cooperative staged load of the 32x128 stripe, fused with row scaling.
  {
    const int r = tid >> 3;                 // 0..31
    const int k = (tid & 7) * 16;           // 0,16,...,112
    const int gr = row0 + r;
    float* q = lA + r * 132 + k;
    if (full || gr < nrows) {
      const float s = rowscale ? rowscale[gr] : 1.0f;
      const float4* p = (const float4*)(A + (size_t)gr * DIM + k);
      const float4 v0 = p[0], v1 = p[1], v2 = p[2], v3 = p[3];
      q[0]  = v0.x * s; q[1]  = v0.y * s; q[2]  = v0.z * s; q[3]  = v0.w * s;
      q[4]  = v1.x * s; q[5]  = v1.y * s; q[6]  = v1.z * s; q[7]  = v1.w * s;
      q[8]  = v2.x * s; q[9]  = v2.y * s; q[10] = v2.z * s; q[11] = v2.w * s;
      q[12] = v3.x * s; q[13] = v3.y * s; q[14] = v3.z * s; q[15] = v3.w * s;
    } else {
      #pragma unroll
      for (int j = 0; j < 16; ++j) q[j] = 0.0f;
    }
  }
  __syncthreads();

  const int colbase = wv * 16;
  const int mn = tx & 15;                   // M for A, N for B/C
  const int kb = (tx >> 4) * 2;             // K sub-offset per half-wave
  const float* lA0 = lA + mn * 132;         // rows 0-15 of stripe
  const float* lA1 = lA + (mn + 16) * 132;  // rows 16-31 of stripe

  v8f c0 = {};
  v8f c1 = {};
  #pragma unroll 4
  for (int k = 0; k < DIM; k += 4) {
    v2f b;
    b.x = W[(size_t)(k + kb) * DIM + colbase + mn];
    b.y = W[(size_t)(k + kb + 1) * DIM + colbase + mn];
    v2f a0, a1;
    a0.x = lA0[k + kb];
    a0.y = lA0[k + kb + 1];
    a1.x = lA1[k + kb];
    a1.y = lA1[k + kb + 1];
    // 8 args: (neg_a, A, neg_b, B, c_mod, C, reuse_a, reuse_b)
    c0 = __builtin_amdgcn_wmma_f32_16x16x4_f32(false, a0, false, b,
                                               (short)0, c0, false, false);
    c1 = __builtin_amdgcn_wmma_f32_16x16x4_f32(false, a1, false, b,
                                               (short)0, c1, false, false);
  }

  const float bv = bias[colbase + mn];
  const int rbase = (tx >> 4) * 8;          // C/D layout: row = i + 8*(lane>=16)
  if (full) {
    // Fast path (6249/6250 stripes): unguarded coalesced stores.
    #pragma unroll
    for (int i = 0; i < 8; ++i) {
      out[(size_t)(row0 + rbase + i) * DIM + colbase + mn] = c0[i] + bv;
      out[(size_t)(row0 + 16 + rbase + i) * DIM + colbase + mn] = c1[i] + bv;
    }
  } else {
    #pragma unroll
    for (int i = 0; i < 8; ++i) {
      const int g0 = row0 + rbase + i;
      if (g0 < nrows) out[(size_t)g0 * DIM + colbase + mn] = c0[i] + bv;
      const int g1 = row0 + 16 + rbase + i;
      if (g1 < nrows) out[(size_t)g1 * DIM + colbase + mn] = c1[i] + bv;
    }
  }
}

// ---------------------------------------------------------------------------
// Edge scatter: agg[dst[e], :] += h[src[e], :] * norm_src[src[e]]
// One thread per (edge, 4-dim chunk): 32 threads per edge, float4 gather,
// 4x global_atomic_add_f32 into the L2-resident agg buffer.
// ---------------------------------------------------------------------------
__global__ void gcn_scatter_kernel(const float* __restrict__ h,
                                   const int* __restrict__ src,
                                   const int* __restrict__ dst,
                                   const float* __restrict__ nsrc,
                                   float* __restrict__ agg, int nedges) {
  int idx = blockIdx.x * blockDim.x + threadIdx.x;
  int e = idx >> 5;
  if (e >= nedges) return;
  int c = (idx & 31) * 4;
  int s = src[e];
  int d = dst[e];
  float sc = nsrc[s];
  const float4 v = *(const float4*)(h + (size_t)s * DIM + c);
  float* o = agg + (size_t)d * DIM + c;
  atomicAdd(o + 0, v.x * sc);
  atomicAdd(o + 1, v.y * sc);
  atomicAdd(o + 2, v.z * sc);
  atomicAdd(o + 3, v.w * sc);
}

// ---------------------------------------------------------------------------
// BN column stats: stats[0..127] = sum, stats[128..255] = sum of squares
// blockDim = 128 (one column per thread), grid-stride over rows.
// ---------------------------------------------------------------------------
__global__ void gcn_bn_stats_kernel(const float* __restrict__ x,
                                    float* __restrict__ stats, int n) {
  const int col = threadIdx.x;
  float s = 0.0f, s2 = 0.0f;
  for (int r = blockIdx.x; r < n; r += gridDim.x) {
    float v = x[(size_t)r * DIM + col];
    s += v;
    s2 += v * v;
  }
  atomicAdd(&stats[col], s);
  atomicAdd(&stats[DIM + col], s2);
}

// BN apply + ReLU + residual:  h = relu((pre-mu)*rsqrt(var+eps)*gamma+beta) + h
__global__ void gcn_bn_apply_kernel(const float* __restrict__ pre,
                                    const float* __restrict__ stats,
                                    const float* __restrict__ gamma,
                                    const float* __restrict__ beta,
                                    float* __restrict__ h, int n) {
  int idx = blockIdx.x * blockDim.x + threadIdx.x;
  if (idx >= n * DIM) return;
  const int col = idx & (DIM - 1);
  const float invN = 1.0f / (float)n;
  const float mu = stats[col] * invN;
  const float var = stats[DIM + col] * invN - mu * mu;
  const float inv = rsqrtf(var + EPS);
  float v = (pre[idx] - mu) * inv * gamma[col] + beta[col];
  v = fmaxf(v, 0.0f);
  h[idx] = v + h[idx];
}

// ---------------------------------------------------------------------------
// Readout: per-graph sums (atomics) then divide by counts.
// ---------------------------------------------------------------------------
__global__ void gcn_readout_acc_kernel(const float* __restrict__ h,
                                       const int* __restrict__ gid,
                                       float* __restrict__ gsum,
                                       float* __restrict__ gcnt, int n) {
  int idx = blockIdx.x * blockDim.x + threadIdx.x;
  int node = idx >> 5;
  if (node >= n) return;
  int c = (idx & 31) * 4;
  int g = gid[node];
  const float4 v = *(const float4*)(h + (size_t)node * DIM + c);
  float* o = gsum + (size_t)g * DIM + c;
  atomicAdd(o + 0, v.x);
  atomicAdd(o + 1, v.y);
  atomicAdd(o + 2, v.z);
  atomicAdd(o + 3, v.w);
  if (c == 0) atomicAdd(&gcnt[g], 1.0f);
}

__global__ void gcn_readout_div_kernel(const float* __restrict__ gsum,
                                       const float* __restrict__ gcnt,
                                       float* __restrict__ out, int total) {
  int idx = blockIdx.x * blockDim.x + threadIdx.x;
  if (idx >= total) return;
  int g = idx >> 7;  // /DIM
  out[idx] = gsum[idx] / fmaxf(gcnt[g], 1.0f);
}

// ---------------------------------------------------------------------------
// Host launcher
// ---------------------------------------------------------------------------
extern "C" void kernel_launch(void* const* d_in, const int* in_sizes, int n_in,
                              void* d_out, int out_size, void* d_ws, size_t ws_size,
                              hipStream_t stream) {
  (void)n_in; (void)ws_size;
  const float* h_in    = (const float*)d_in[0];
  const int*   src     = (const int*)d_in[1];
  const int*   dst     = (const int*)d_in[2];
  const int*   gid     = (const int*)d_in[3];
  const float* W_embed = (const float*)d_in[4];
  const float* b_embed = (const float*)d_in[5];
  const float* W_lay   = (const float*)d_in[6];
  const float* b_lay   = (const float*)d_in[7];
  const float* gamma   = (const float*)d_in[8];
  const float* beta    = (const float*)d_in[9];

  const int N = in_sizes[0] / DIM;
  const int E = in_sizes[1];
  const int NLAYERS = in_sizes[6] / (DIM * DIM);
  const int NGRAPHS = out_size / DIM;

  // Workspace carving (256B aligned)
  auto align_up = [](size_t v) { return (v + 255) & ~(size_t)255; };
  char* ws = (char*)d_ws;
  size_t off = 0;
  float* h_cur = (float*)(ws + off); off = align_up(off + (size_t)N * DIM * 4);
  float* buf   = (float*)(ws + off); off = align_up(off + (size_t)N * DIM * 4);
  float* nsrc  = (float*)(ws + off); off = align_up(off + (size_t)N * 4);
  float* ndst  = (float*)(ws + off); off = align_up(off + (size_t)N * 4);
  float* stats = (float*)(ws + off); off = align_up(off + 2 * DIM * 4);
  float* gsum  = (float*)(ws + off); off = align_up(off + (size_t)NGRAPHS * DIM * 4);
  float* gcnt  = (float*)(ws + off); off = align_up(off + (size_t)NGRAPHS * 4);

  const int nstripes = (N + 31) / 32;
  dim3 gemm_grid(nstripes);
  dim3 gemm_block(32, 8);

  // 1) degrees -> norms
  hipMemsetAsync(nsrc, 0, (size_t)N * 4, stream);
  hipMemsetAsync(ndst, 0, (size_t)N * 4, stream);
  gcn_degree_kernel<<<(E + 255) / 256, 256, 0, stream>>>(src, dst, nsrc, ndst, E);
  gcn_norm_kernel<<<(N + 255) / 256, 256, 0, stream>>>(nsrc, N);
  gcn_norm_kernel<<<(N + 255) / 256, 256, 0, stream>>>(ndst, N);

  // 2) input embedding: h_cur = h_in @ W_embed + b_embed
  gcn_gemm_wmma<<<gemm_grid, gemm_block, 0, stream>>>(
      h_in, W_embed, b_embed, (const float*)nullptr, h_cur, N);

  // 3) GCN layers
  const long scatter_threads = (long)E * 32;
  const int scatter_blocks = (int)((scatter_threads + 255) / 256);
  for (int l = 0; l < NLAYERS; ++l) {
    hipMemsetAsync(buf, 0, (size_t)N * DIM * 4, stream);
    gcn_scatter_kernel<<<scatter_blocks, 256, 0, stream>>>(h_cur, src, dst, nsrc, buf, E);
    // buf = (buf * ndst) @ W_l + b_l   (in place per 32-row stripe)
    gcn_gemm_wmma<<<gemm_grid, gemm_block, 0, stream>>>(
        buf, W_lay + (size_t)l * DIM * DIM, b_lay + (size_t)l * DIM, ndst, buf, N);
    hipMemsetAsync(stats, 0, 2 * DIM * 4, stream);
    gcn_bn_stats_kernel<<<512, 128, 0, stream>>>(buf, stats, N);
    gcn_bn_apply_kernel<<<((long)N * DIM + 255) / 256, 256, 0, stream>>>(
        buf, stats, gamma + (size_t)l * DIM, beta + (size_t)l * DIM, h_cur, N);
  }

  // 4) mean readout per graph
  hipMemsetAsync(gsum, 0, (size_t)NGRAPHS * DIM * 4, stream);
  hipMemsetAsync(gcnt, 0, (size_t)NGRAPHS * 4, stream);
  gcn_readout_acc_kernel<<<(int)(((long)N * 32 + 255) / 256), 256, 0, stream>>>(
      h_cur, gid, gsum, gcnt, N);
  gcn_readout_div_kernel<<<(NGRAPHS * DIM + 255) / 256, 256, 0, stream>>>(
      gsum, gcnt, (float*)d_out, NGRAPHS * DIM);
}